// sparse_att_layer_68753836474510
// MI455X (gfx1250) — compile-verified
//
#include <hip/hip_runtime.h>
#include <math.h>

typedef __attribute__((ext_vector_type(2))) float v2f;
typedef __attribute__((ext_vector_type(8))) float v8f;

#define DIM 128

// ---------------------------------------------------------------------------
// Kernel 0: zero output + denominator slot
// ---------------------------------------------------------------------------
__global__ void zero_kernel(float4* __restrict__ out4, long long n4,
                            float* __restrict__ denom) {
    long long i = (long long)blockIdx.x * blockDim.x + threadIdx.x;
    if (i == 0) *denom = 0.0f;
    long long stride = (long long)gridDim.x * blockDim.x;
    float4 z = make_float4(0.f, 0.f, 0.f, 0.f);
    for (; i < n4; i += stride) out4[i] = z;
}

// ---------------------------------------------------------------------------
// Kernel 1 (single block, 1024 threads = 32 waves):
//   scores = dual_layer @ conv_w + b   via V_WMMA_F32_16X16X4_F32 (exact f32)
//   M = max_r lrelu(score_r);  p[r] = exp(lrelu(score_r) - M)  -> ws p table
// B-matrix trick: all 16 columns hold the same conv_w chunk, so every column
// of D equals the 16 partial scores; extract column 0 (lanes 0 and 16).
// ---------------------------------------------------------------------------
__global__ __launch_bounds__(1024)
void scores_kernel(const float* __restrict__ dual,
                   const float* __restrict__ cw,
                   const float* __restrict__ cb,
                   float* __restrict__ wsF, int nrel) {
    __shared__ float s_lr[1024];   // lrelu(scores), padded
    __shared__ float s_red[32];
    const int tid  = threadIdx.x;
    const int lane = tid & 31;
    const int wave = tid >> 5;
    const int ntiles = (nrel + 15) >> 4;

    const int half = lane >> 4;          // 0: K=0,1   1: K=2,3  (A & B halves)
    const int m    = lane & 15;

    for (int t = wave; t < ntiles; t += 32) {
        const int base = t * 16;
        int row = base + m;
        if (row >= nrel) row = nrel - 1;           // clamp (discarded later)
        const float* arow = dual + (long long)row * DIM;

        v8f acc = {};
        for (int k = 0; k < DIM; k += 4) {
            const int kk = k + half * 2;
            v2f a, b;
            a.x = arow[kk];   a.y = arow[kk + 1];
            b.x = cw[kk];     b.y = cw[kk + 1];    // broadcast into all 16 cols
            acc = __builtin_amdgcn_wmma_f32_16x16x4_f32(
                      false, a, false, b, (short)0, acc, false, false);
        }
        // C/D layout: VGPR v -> lanes 0-15: M=v ; lanes 16-31: M=v+8 ; N=lane%16
        if (lane == 0 || lane == 16) {
            const int moff = (lane == 16) ? 8 : 0;
            #pragma unroll
            for (int i = 0; i < 8; ++i) {
                const int r = base + moff + i;
                if (r < nrel) {
                    float sc = acc[i] + cb[0];
                    s_lr[r] = (sc > 0.f) ? sc : 0.01f * sc;   // leaky_relu
                }
            }
        }
    }
    __syncthreads();

    // block max over s_lr[0..nrel)
    float mx = -INFINITY;
    for (int r = tid; r < nrel; r += 1024) mx = fmaxf(mx, s_lr[r]);
    #pragma unroll
    for (int off = 16; off; off >>= 1) mx = fmaxf(mx, __shfl_xor(mx, off, 32));
    if (lane == 0) s_red[wave] = mx;
    __syncthreads();
    if (wave == 0) {
        float v = s_red[lane];
        #pragma unroll
        for (int off = 16; off; off >>= 1) v = fmaxf(v, __shfl_xor(v, off, 32));
        if (lane == 0) s_red[0] = v;
    }
    __syncthreads();
    const float M = s_red[0];
    for (int r = tid; r < nrel; r += 1024)
        wsF[8 + r] = expf(s_lr[r] - M);            // p table
}

// ---------------------------------------------------------------------------
// Kernel 2: denom = sum_e p[rel_e]   (p staged in LDS, wave32 reduce, 1 atomic)
// ---------------------------------------------------------------------------
__global__ __launch_bounds__(256)
void denom_kernel(const int* __restrict__ rel, long long nnz,
                  const float* __restrict__ p_glob, float* __restrict__ denom,
                  int nrel) {
    __shared__ float p_s[1024];
    __shared__ float red[8];
    for (int i = threadIdx.x; i < nrel; i += blockDim.x) p_s[i] = p_glob[i];
    __syncthreads();

    float sum = 0.f;
    const long long stride = (long long)gridDim.x * blockDim.x;
    for (long long e = (long long)blockIdx.x * blockDim.x + threadIdx.x;
         e < nnz; e += stride)
        sum += p_s[rel[e]];

    const int lane = threadIdx.x & 31, wave = threadIdx.x >> 5;
    #pragma unroll
    for (int off = 16; off; off >>= 1) sum += __shfl_xor(sum, off, 32);
    if (lane == 0) red[wave] = sum;
    __syncthreads();
    if (threadIdx.x == 0) {
        float t = 0.f;
        #pragma unroll
        for (int i = 0; i < 8; ++i) t += red[i];
        atomicAdd(denom, t);
    }
}

// ---------------------------------------------------------------------------
// Kernel 3: scatter  out[row_e] += (p[rel_e]/denom) * inlayer[col_e]
// One wave per 32-edge batch; lanes cooperatively move one 512B row at a time.
// ---------------------------------------------------------------------------
__global__ __launch_bounds__(256)
void scatter_kernel(const float* __restrict__ inlayer,
                    const int* __restrict__ rowi,
                    const int* __restrict__ coli,
                    const int* __restrict__ rel,
                    const float* __restrict__ wsF,
                    float* __restrict__ out, long long nnz) {
    const float inv = 1.0f / wsF[0];
    const float* p = wsF + 8;
    const int lane = threadIdx.x & 31;
    const long long gw = ((long long)blockIdx.x * blockDim.x + threadIdx.x) >> 5;
    const long long nw = ((long long)gridDim.x * blockDim.x) >> 5;

    for (long long base = gw * 32; base < nnz; base += nw * 32) {
        const long long e = base + lane;
        int r = 0, c = 0; float coef = 0.f;
        if (e < nnz) {
            r = rowi[e];
            c = coli[e];
            coef = p[rel[e]] * inv;
            // warm L2 for this wave's 32 gathered rows (global_prefetch_b8)
            __builtin_prefetch(inlayer + (long long)c * DIM, 0, 1);
        }
        const int cnt = (int)((nnz - base < 32) ? (nnz - base) : 32);
        for (int j = 0; j < cnt; ++j) {
            const int   cj = __shfl(c, j, 32);
            const int   rj = __shfl(r, j, 32);
            const float kf = __shfl(coef, j, 32);
            const float4 v = *(const float4*)(inlayer + (long long)cj * DIM + lane * 4);
            float* op = out + (long long)rj * DIM + lane * 4;
            atomicAdd(op + 0, kf * v.x);
            atomicAdd(op + 1, kf * v.y);
            atomicAdd(op + 2, kf * v.z);
            atomicAdd(op + 3, kf * v.w);
        }
    }
}

// ---------------------------------------------------------------------------
extern "C" void kernel_launch(void* const* d_in, const int* in_sizes, int n_in,
                              void* d_out, int out_size, void* d_ws, size_t ws_size,
                              hipStream_t stream) {
    const float* inlayer = (const float*)d_in[0];
    const float* dual    = (const float*)d_in[1];
    const float* conv_w  = (const float*)d_in[2];
    const float* conv_b  = (const float*)d_in[3];
    const int*   eidx    = (const int*)d_in[4];
    const int*   erel    = (const int*)d_in[5];
    float*       out     = (float*)d_out;
    float*       wsF     = (float*)d_ws;    // [0]=denom, [8..8+nrel)=p table

    const long long nnz  = in_sizes[5];
    const int       nrel = in_sizes[1] / DIM;
    const int*      rowi = eidx;
    const int*      coli = eidx + nnz;

    // 0) zero out + denom
    const long long n4 = (long long)out_size / 4;
    int zb = (int)((n4 + 255) / 256); if (zb > 6256) zb = 6256;
    zero_kernel<<<zb, 256, 0, stream>>>((float4*)out, n4, wsF);

    // 1) scores -> lrelu -> max -> p table (WMMA f32 16x16x4)
    scores_kernel<<<1, 1024, 0, stream>>>(dual, conv_w, conv_b, wsF, nrel);

    // 2) denom reduction over edges
    int db = (int)((nnz + 255) / 256); if (db > 2048) db = 2048;
    denom_kernel<<<db, 256, 0, stream>>>(erel, nnz, wsF + 8, wsF, nrel);

    // 3) scatter-accumulate: one wave per 32-edge batch
    int sb = (int)((nnz + 255) / 256); if (sb < 1) sb = 1;
    scatter_kernel<<<sb, 256, 0, stream>>>(inlayer, rowi, coli, erel, wsF, out, nnz);
}